// EquivariantGraphConv_28321014350244
// MI455X (gfx1250) — compile-verified
//
#include <hip/hip_runtime.h>

typedef float v2f __attribute__((ext_vector_type(2)));
typedef float v8f __attribute__((ext_vector_type(8)));

#define N_NODES 10000
#define N_EDGES 160000
#define HIDDEN  128
#define FEAT    512   // 4*HIDDEN floats per node row (scalar | 3x vector)

// ---------------------------------------------------------------------------
// Fused per-node linear transform, fp32 WMMA 16x16x4.
//   rel[n, cols] = in[n] @ Wrel.T          (message term -> workspace)
//   out[n, cols] = in[n] @ Wroot.T (+bias) (root term    -> d_out)
//
// Template KDIM (compile-time) keeps the K loop a scalar-counted loop, so the
// WMMAs run with EXEC all-ones and the loop unrolls cleanly.
// Each wave owns a 16(node) x 32(col) tile: one A operand feeds 4 independent
// WMMA accumulator chains (rel/root x 2 column tiles).
//   SCALAR pass:  4 waves x 32 cols = 128 outputs, KDIM=128
//   VECTOR pass: 12 waves x 32 cols = 384 outputs, KDIM=384
// ---------------------------------------------------------------------------
template <int KDIM, bool SCALAR>
__global__ void node_transform_kernel(const float* __restrict__ x,
                                      const float* __restrict__ Wrel,
                                      const float* __restrict__ Wroot,
                                      const float* __restrict__ bs,
                                      float* __restrict__ out,
                                      float* __restrict__ rel)
{
    const int wave = threadIdx.x >> 5;
    const int lane = threadIdx.x & 31;
    const int half = lane >> 4;          // which 16-lane half of the wave
    const int m    = lane & 15;

    const int node0 = blockIdx.x * 16;
    const int j0    = wave * 32;         // first of two 16-wide column tiles

    constexpr int COL_OFF = SCALAR ? 0 : HIDDEN;   // position in 512-wide row
    const float* Abase = x + (size_t)node0 * FEAT + COL_OFF;

    // A-tile: lane (half,m) supplies A[m, k+2*half .. +1]
    // B-tile: lane (half,m) supplies W[jrow+m, k+2*half .. +1]
    const float* arow = Abase + (size_t)m * FEAT;
    const float* wr0  = Wrel  + (size_t)(j0 + m)      * KDIM;   // rel,  cols j0..j0+15
    const float* wr1  = Wrel  + (size_t)(j0 + 16 + m) * KDIM;   // rel,  cols j0+16..j0+31
    const float* wt0  = Wroot + (size_t)(j0 + m)      * KDIM;   // root, tile 0
    const float* wt1  = Wroot + (size_t)(j0 + 16 + m) * KDIM;   // root, tile 1
    const int koff = 2 * half;

    v8f accR0 = {0.f,0.f,0.f,0.f,0.f,0.f,0.f,0.f};
    v8f accR1 = accR0, accT0 = accR0, accT1 = accR0;

#pragma unroll 4
    for (int k = 0; k < KDIM; k += 4) {
        const int k2 = k + koff;
        v2f a  = { arow[k2], arow[k2 + 1] };
        v2f b0 = { wr0[k2],  wr0[k2 + 1]  };
        v2f b1 = { wr1[k2],  wr1[k2 + 1]  };
        v2f b2 = { wt0[k2],  wt0[k2 + 1]  };
        v2f b3 = { wt1[k2],  wt1[k2 + 1]  };
        // Four independent accumulator chains share the A operand.
        accR0 = __builtin_amdgcn_wmma_f32_16x16x4_f32(false, a, false, b0, (short)0, accR0, false, false);
        accR1 = __builtin_amdgcn_wmma_f32_16x16x4_f32(false, a, false, b1, (short)0, accR1, false, false);
        accT0 = __builtin_amdgcn_wmma_f32_16x16x4_f32(false, a, false, b2, (short)0, accT0, false, false);
        accT1 = __builtin_amdgcn_wmma_f32_16x16x4_f32(false, a, false, b3, (short)0, accT1, false, false);
    }

    float bias0 = 0.0f, bias1 = 0.0f;
    if constexpr (SCALAR) {              // root bias exists only on scalar part
        bias0 = bs[j0 + m];
        bias1 = bs[j0 + 16 + m];
    }

    // D layout: accumulator element r is row M = r + 8*half, col N = m.
#pragma unroll
    for (int r = 0; r < 8; ++r) {
        const int    node = node0 + r + 8 * half;
        const size_t off  = (size_t)node * FEAT + COL_OFF + j0 + m;
        rel[off]      = accR0[r];
        rel[off + 16] = accR1[r];
        out[off]      = accT0[r] + bias0;   // root term seeds the output
        out[off + 16] = accT1[r] + bias1;
    }
}

// ---------------------------------------------------------------------------
// Edge scatter: out[row] += rel[col], 512 f32 per edge.
// One 256-thread block per edge; edge indices are blockIdx-uniform (scalar
// loads), float2 coalesced reads from the L2-resident 20 MB workspace, f32
// global atomic adds. This is the HBM/L2 roofline of the whole op.
// ---------------------------------------------------------------------------
__global__ __launch_bounds__(256)
void edge_scatter_kernel(const int* __restrict__ edge,
                         const float* __restrict__ rel,
                         float* __restrict__ out)
{
    const int e   = blockIdx.x;
    const int dst = edge[e];             // edge_index[0][e]  (segment / row)
    const int src = edge[N_EDGES + e];   // edge_index[1][e]  (source  / col)
    const int t   = threadIdx.x;

    const float2* s = (const float2*)(rel + (size_t)src * FEAT);
    const float2  v = s[t];

    float* d = out + (size_t)dst * FEAT + 2 * t;
    atomicAdd(d,     v.x);
    atomicAdd(d + 1, v.y);
}

extern "C" void kernel_launch(void* const* d_in, const int* in_sizes, int n_in,
                              void* d_out, int out_size, void* d_ws, size_t ws_size,
                              hipStream_t stream)
{
    const float* x    = (const float*)d_in[0];
    const int*   edge = (const int*)  d_in[1];
    const float* Wsr  = (const float*)d_in[2];
    const float* Wsro = (const float*)d_in[3];
    const float* bs   = (const float*)d_in[4];
    const float* Wvr  = (const float*)d_in[5];
    const float* Wvro = (const float*)d_in[6];

    float* out = (float*)d_out;
    float* rel = (float*)d_ws;   // N_NODES * FEAT floats = 20.48 MB workspace

    const int ntiles = N_NODES / 16;     // 625, exact

    // Scalar block:  out[:,0:128],   K=128, 4 waves x 32 cols
    node_transform_kernel<HIDDEN, true>
        <<<ntiles, 128, 0, stream>>>(x, Wsr, Wsro, bs, out, rel);

    // Vector block:  out[:,128:512], K=384, 12 waves x 32 cols
    node_transform_kernel<3 * HIDDEN, false>
        <<<ntiles, 384, 0, stream>>>(x, Wvr, Wvro, nullptr, out, rel);

    // Message aggregation (segment_sum over edges)
    edge_scatter_kernel<<<N_EDGES, 256, 0, stream>>>(edge, rel, out);
}